// BucketingBBoxCoderWraper_1202590843769
// MI455X (gfx1250) — compile-verified
//
#include <hip/hip_runtime.h>
#include <stdint.h>

#define TPB 256
#define NSIDE 7   // buckets per side (SIDE = ceil(14/2) in reference)

typedef float fx4 __attribute__((ext_vector_type(4)));   // native vector: OK for nontemporal builtins

// ---- CDNA5 (gfx1250) async global->LDS streaming copy helpers ----
// ISA §10.6/§15.18: GLOBAL_LOAD_ASYNC_TO_LDS_B128, tracked by ASYNCcnt.
// INST_OFFSET is added to BOTH the LDS address (VDST) and global address (VADDR),
// so a single immediate advances the whole coalesced copy window.
// th:TH_LOAD_NT -> data is streamed once; don't displace useful L2 lines.
template<int OFF>
__device__ __forceinline__ void async_ld_b128(uint32_t lds_byte_addr, const float* gptr) {
  asm volatile("global_load_async_to_lds_b128 %0, %1, off offset:%2 th:TH_LOAD_NT"
               :: "v"(lds_byte_addr), "v"(gptr), "i"(OFF)
               : "memory");
}
__device__ __forceinline__ void async_ld_b128_dyn(uint32_t lds_byte_addr, const float* gptr) {
  asm volatile("global_load_async_to_lds_b128 %0, %1, off th:TH_LOAD_NT"
               :: "v"(lds_byte_addr), "v"(gptr)
               : "memory");
}
__device__ __forceinline__ void wait_asynccnt0() {
  asm volatile("s_wait_asynccnt 0x0" ::: "memory");
}

__global__ __launch_bounds__(TPB)
void bucket_coder_kernel(const float* __restrict__ prop,
                         const float* __restrict__ cls,
                         const float* __restrict__ offp,
                         float* __restrict__ bbox_out,
                         float* __restrict__ conf_out,
                         int P)
{
  // 256 proposals/block * 28 floats = 28 KB per array; 56 KB total (<< 320 KB/WGP)
  __shared__ fx4 s_cls[TPB * NSIDE];
  __shared__ fx4 s_off[TPB * NSIDE];

  const int t    = threadIdx.x;
  const int base = blockIdx.x * TPB;
  const int i    = base + t;

  // Low 32 bits of a generic pointer to LDS == LDS byte offset.
  const uint32_t lds_c = (uint32_t)(uintptr_t)&s_cls[0] + (uint32_t)t * 16u;
  const uint32_t lds_o = (uint32_t)(uintptr_t)&s_off[0] + (uint32_t)t * 16u;
  const float* gc = cls  + ((long long)base * NSIDE + t) * 4;  // per-thread float4 base
  const float* go = offp + ((long long)base * NSIDE + t) * 4;

  if (base + TPB <= P) {
    // Full block (common case): 14 straight-line async b128 copies, fully coalesced,
    // immediate offsets advance LDS+global windows together. No branches, no VALU.
    async_ld_b128<0 * 4096>(lds_c, gc);  async_ld_b128<0 * 4096>(lds_o, go);
    async_ld_b128<1 * 4096>(lds_c, gc);  async_ld_b128<1 * 4096>(lds_o, go);
    async_ld_b128<2 * 4096>(lds_c, gc);  async_ld_b128<2 * 4096>(lds_o, go);
    async_ld_b128<3 * 4096>(lds_c, gc);  async_ld_b128<3 * 4096>(lds_o, go);
    async_ld_b128<4 * 4096>(lds_c, gc);  async_ld_b128<4 * 4096>(lds_o, go);
    async_ld_b128<5 * 4096>(lds_c, gc);  async_ld_b128<5 * 4096>(lds_o, go);
    async_ld_b128<6 * 4096>(lds_c, gc);  async_ld_b128<6 * 4096>(lds_o, go);
  } else {
    // Rare tail block: per-element guard.
    const long long tot4 = (long long)P * NSIDE;
    for (int j = 0; j < NSIDE; ++j) {
      const long long e = (long long)base * NSIDE + j * TPB + t;
      if (e < tot4) {
        async_ld_b128_dyn(lds_c + (uint32_t)j * 4096u, cls  + e * 4);
        async_ld_b128_dyn(lds_o + (uint32_t)j * 4096u, offp + e * 4);
      }
    }
  }
  wait_asynccnt0();   // each wave drains its own ASYNCcnt ...
  __syncthreads();    // ... then cross-wave visibility

  if (i >= P) return;

  const fx4 p = __builtin_nontemporal_load((const fx4*)prop + i);  // coalesced, streamed once
  const float cx = (p.x + p.z) * 0.5f;
  const float cy = (p.y + p.w) * 0.5f;
  const float pw = (p.z - p.x) * 3.0f;    // SCALE_FACTOR
  const float ph = (p.w - p.y) * 3.0f;
  const float px1 = cx - 0.5f * pw, px2 = cx + 0.5f * pw;
  const float py1 = cy - 0.5f * ph, py2 = cy + 0.5f * ph;
  const float bw = pw * (1.0f / 14.0f);   // / NUM_BUCKETS
  const float bh = ph * (1.0f / 14.0f);

  // Pull this thread's 28+28 floats out of LDS (7+7 x ds_load_b128, 16B aligned).
  float cl[28], of[28];
#pragma unroll
  for (int j = 0; j < NSIDE; ++j) {
    const fx4 a = s_cls[t * NSIDE + j];
    cl[4*j+0]=a.x; cl[4*j+1]=a.y; cl[4*j+2]=a.z; cl[4*j+3]=a.w;
    const fx4 b = s_off[t * NSIDE + j];
    of[4*j+0]=b.x; of[4*j+1]=b.y; of[4*j+2]=b.z; of[4*j+3]=b.w;
  }

  // Sides s=0..3 => l, r, t, d. Layout: i*28 + s*7 + k for both arrays.
  float ind[4], ov[4];
  float conf = 0.0f;
#pragma unroll
  for (int s = 0; s < 4; ++s) {
    // Single scan: top-2 of logits (== top-2 of softmax, monotone) with fused
    // offset gather at the running top-1. Strict '>' preserves jax.lax.top_k
    // tie-breaking (lowest index wins). Indices tracked as floats (their only
    // consumers are float expressions) to avoid v_cvt_f32_i32.
    float m1 = cl[s*7+0]; float fi1 = 0.0f; float o1 = of[s*7+0];
    float m2 = -3.402823466e38f; float fi2 = 0.0f;
#pragma unroll
    for (int k = 1; k < 7; ++k) {
      const float v  = cl[s*7+k];
      const float fk = (float)k;
      if (v > m1)      { m2 = m1; fi2 = fi1; m1 = v; fi1 = fk; o1 = of[s*7+k]; }
      else if (v > m2) { m2 = v; fi2 = fk; }
    }
    float sum = 0.0f;
#pragma unroll
    for (int k = 0; k < 7; ++k) sum += __expf(cl[s*7+k] - m1);
    // sum in [1, 7]: no denorm/inf/NaN edge cases -> raw v_rcp_f32 (~1 ulp) is safe
    // and replaces the ~10-instruction IEEE divide expansion.
    const float inv = __builtin_amdgcn_rcpf(sum);
    const float s1  = inv;                      // exp(m1-m1)/sum
    const float s2  = __expf(m2 - m1) * inv;
    conf += s1 + s2 * (fabsf(fi1 - fi2) - 1.0f);
    ind[s] = fi1;
    ov[s]  = o1;
  }

  float x1 = px1 + (0.5f + ind[0] - ov[0]) * bw;
  float x2 = px2 - (0.5f + ind[1] + ov[1]) * bw;
  float y1 = py1 + (0.5f + ind[2] - ov[2]) * bh;
  float y2 = py2 - (0.5f + ind[3] + ov[3]) * bh;
  x1 = fminf(fmaxf(x1, 0.0f), 1332.0f);   // W-1
  x2 = fminf(fmaxf(x2, 0.0f), 1332.0f);
  y1 = fminf(fmaxf(y1, 0.0f),  799.0f);   // H-1
  y2 = fminf(fmaxf(y2, 0.0f),  799.0f);

  const fx4 box = { x1, y1, x2, y2 };
  __builtin_nontemporal_store(box, (fx4*)bbox_out + i);
  __builtin_nontemporal_store(conf * 0.25f, conf_out + i);
}

extern "C" void kernel_launch(void* const* d_in, const int* in_sizes, int n_in,
                              void* d_out, int out_size, void* d_ws, size_t ws_size,
                              hipStream_t stream) {
  const float* prop = (const float*)d_in[0];   // (B, N, 4) f32
  const float* cls  = (const float*)d_in[1];   // (B, N*4*7) f32
  const float* offp = (const float*)d_in[2];   // (B, N*4*7) f32
  const int P = in_sizes[0] / 4;               // B*N proposals
  float* out  = (float*)d_out;                 // [bboxes | loc_conf] flat
  float* bbox = out;
  float* conf = out + (size_t)P * 4;
  const int grid = (P + TPB - 1) / TPB;
  bucket_coder_kernel<<<grid, TPB, 0, stream>>>(prop, cls, offp, bbox, conf, P);
  (void)n_in; (void)out_size; (void)d_ws; (void)ws_size;
}